// DecoderLayer_4595615007474
// MI455X (gfx1250) — compile-verified
//
#include <hip/hip_runtime.h>
#include <hip/hip_bf16.h>

typedef __attribute__((ext_vector_type(2))) float v2f;
typedef __attribute__((ext_vector_type(8))) float v8f;

#define BB 8
#define LL 4096
#define HH 32
#define DD 128
#define HID 4096
#define INTER 11008
#define SPLITS 4
#define LC (LL / SPLITS)

// ---- workspace layout (float offsets) ----
#define OFF_HNORM 0          // 16*4096        = 65536
#define OFF_QKV   65536      // 16*12288       = 196608
#define OFF_QR    262144     // 8*32*128       = 32768
#define OFF_KN    294912     // 32768
#define OFF_VN    327680     // 32768
#define OFF_PART  360448     // 8*32*4*130     = 133120
#define OFF_CTX   493568     // 16*4096        = 65536
#define OFF_H1    559104     // 8*4096         = 32768
#define OFF_H2    591872     // 16*4096        = 65536
#define OFF_ACT   657408     // 16*11008       = 176128
#define OFF_ARENA 833536     // split-K partial arena (max 2*16*11008*2 = 704512)
// total ws: 1538048 floats ~= 6.2 MB

// ---------------- RMSNorm into 16-row zero-padded buffer ----------------
__global__ void rmsnorm16_kernel(const float* __restrict__ x, const float* __restrict__ w,
                                 float* __restrict__ out_pad) {
  const int b = blockIdx.x;        // 0..15
  const int t = threadIdx.x;       // 256
  __shared__ float red[256];
  if (b >= BB) {
    for (int k = t; k < HID; k += 256) out_pad[(size_t)b * HID + k] = 0.f;
    return;
  }
  const float* xr = x + (size_t)b * HID;
  float ss = 0.f;
  for (int k = t; k < HID; k += 256) { float v = xr[k]; ss += v * v; }
  red[t] = ss; __syncthreads();
  for (int o = 128; o > 0; o >>= 1) { if (t < o) red[t] += red[t + o]; __syncthreads(); }
  const float rs = rsqrtf(red[0] / (float)HID + 1e-6f);
  for (int k = t; k < HID; k += 256) out_pad[(size_t)b * HID + k] = w[k] * xr[k] * rs;
}

// ---------------- WMMA f32 16x16x4 ranged tile accumulator ----------------
// A: 16 x K (row-major, 16 padded rows), W: N x K row-major (16 rows used as B^T)
// lane layout (ISA 7.12.2): row = lane&15, k = 2*(lane>>4) + j  -> contiguous K-pair
__device__ __forceinline__ v8f gemm_tile_acc_range(const float* __restrict__ A,
                                                   const float* __restrict__ W,
                                                   int n0, int K, int k0, int klen) {
  const int lane  = threadIdx.x & 31;
  const int col   = lane & 15;
  const int khalf = (lane >> 4) << 1;
  const float* ap = A + (size_t)col * K + k0 + khalf;
  const float* wp = W + (size_t)(n0 + col) * K + k0 + khalf;
  v8f c = {};
#pragma unroll 8
  for (int k = 0; k < klen; k += 4) {
    v2f a = *reinterpret_cast<const v2f*>(ap + k);
    v2f b = *reinterpret_cast<const v2f*>(wp + k);
    c = __builtin_amdgcn_wmma_f32_16x16x4_f32(false, a, false, b, (short)0, c, false, false);
  }
  return c;
}

// Split-K GEMM: wave (tile,chunk) accumulates K/SK slice, stores 16-row partial tile.
// Part layout: [SK][16][N]
__global__ void gemm16_splitk_kernel(const float* __restrict__ A, const float* __restrict__ W,
                                     float* __restrict__ Part, int K, int N, int SK) {
  const int wid   = blockIdx.x * (blockDim.x >> 5) + (threadIdx.x >> 5);
  const int chunk = wid % SK;
  const int n0    = (wid / SK) * 16;
  const int klen  = K / SK;
  v8f c = gemm_tile_acc_range(A, W, n0, K, chunk * klen, klen);
  const int lane = threadIdx.x & 31;
  const int mb = (lane >> 4) * 8;
  const int n = n0 + (lane & 15);
#pragma unroll
  for (int j = 0; j < 8; ++j)
    Part[((size_t)chunk * 16 + mb + j) * N + n] = c[j];
}

// Fused gate+up split-K: shares the A fragment between the two weight streams.
__global__ void gemm16_gateup_splitk_kernel(const float* __restrict__ A,
                                            const float* __restrict__ Wg,
                                            const float* __restrict__ Wu,
                                            float* __restrict__ GP, float* __restrict__ UP,
                                            int K, int N, int SK) {
  const int wid   = blockIdx.x * (blockDim.x >> 5) + (threadIdx.x >> 5);
  const int chunk = wid % SK;
  const int n0    = (wid / SK) * 16;
  const int klen  = K / SK;
  const int k0    = chunk * klen;
  const int lane  = threadIdx.x & 31;
  const int col   = lane & 15;
  const int khalf = (lane >> 4) << 1;
  const float* ap = A  + (size_t)col * K + k0 + khalf;
  const float* gp = Wg + (size_t)(n0 + col) * K + k0 + khalf;
  const float* up = Wu + (size_t)(n0 + col) * K + k0 + khalf;
  v8f cg = {}, cu = {};
#pragma unroll 4
  for (int k = 0; k < klen; k += 4) {
    v2f a  = *reinterpret_cast<const v2f*>(ap + k);
    v2f bg = *reinterpret_cast<const v2f*>(gp + k);
    v2f bu = *reinterpret_cast<const v2f*>(up + k);
    cg = __builtin_amdgcn_wmma_f32_16x16x4_f32(false, a, false, bg, (short)0, cg, false, false);
    cu = __builtin_amdgcn_wmma_f32_16x16x4_f32(false, a, false, bu, (short)0, cu, false, false);
  }
  const int mb = (lane >> 4) * 8;
  const int n = n0 + col;
#pragma unroll
  for (int j = 0; j < 8; ++j) {
    GP[((size_t)chunk * 16 + mb + j) * N + n] = cg[j];
    UP[((size_t)chunk * 16 + mb + j) * N + n] = cu[j];
  }
}

// ---------------- split-K reduction kernels (deterministic) ----------------
__global__ void reduce_plain_kernel(const float* __restrict__ Part, float* __restrict__ Out,
                                    int N, int SK) {           // elems = 16*N
  const size_t idx = (size_t)blockIdx.x * 256 + threadIdx.x;
  const size_t stride = (size_t)16 * N;
  float s = 0.f;
  for (int c = 0; c < SK; ++c) s += Part[c * stride + idx];
  Out[idx] = s;
}

__global__ void reduce_resid_kernel(const float* __restrict__ Part, const float* __restrict__ resid,
                                    float* __restrict__ Out, int N, int SK) {  // elems = 8*N
  const size_t idx = (size_t)blockIdx.x * 256 + threadIdx.x;
  const int m = (int)(idx / N), n = (int)(idx % N);
  float s = resid[idx];
  for (int c = 0; c < SK; ++c) s += Part[((size_t)c * 16 + m) * N + n];
  Out[idx] = s;
}

__global__ void reduce_gateup_kernel(const float* __restrict__ GP, const float* __restrict__ UP,
                                     float* __restrict__ Out, int N, int SK) { // elems = 16*N
  const size_t idx = (size_t)blockIdx.x * 256 + threadIdx.x;
  const size_t stride = (size_t)16 * N;
  float g = 0.f, u = 0.f;
  for (int c = 0; c < SK; ++c) { g += GP[c * stride + idx]; u += UP[c * stride + idx]; }
  const float s = g / (1.f + __expf(-g));      // silu
  Out[idx] = s * u;                            // pad rows stay 0
}

// ---------------- RoPE split ----------------
__global__ void rope_kernel(const float* __restrict__ qkv_pad, const int* __restrict__ pos,
                            float* __restrict__ qr, float* __restrict__ kn,
                            float* __restrict__ vn) {
  const int bh = blockIdx.x;            // B*H
  const int b = bh >> 5, h = bh & 31;
  const int d = threadIdx.x;            // 128
  const float* base = qkv_pad + (size_t)b * (3 * HID) + h * DD;
  const float qv = base[d];
  const float kv = base[HID + d];
  const float vv = base[2 * HID + d];
  const float p = (float)pos[b];
  const int fi = d & 63;
  const float freq = __powf(10000.f, -(float)fi / 64.f);
  const float a = p * freq;
  const float ca = __cosf(a), sa = __sinf(a);
  float qo, ko;
  if (d < 64) {
    qo = qv * ca - base[d + 64] * sa;
    ko = kv * ca - base[HID + d + 64] * sa;
  } else {
    qo = base[d - 64] * sa + qv * ca;
    ko = base[HID + d - 64] * sa + kv * ca;
  }
  qr[(size_t)bh * DD + d] = qo;
  kn[(size_t)bh * DD + d] = ko;
  vn[(size_t)bh * DD + d] = vv;
}

// ---------------- attention: split-L partials (flash-decoding style) ----------------
__global__ void attn_partial_kernel(const float* __restrict__ q, const float* __restrict__ kc,
                                    const float* __restrict__ vc, const float* __restrict__ kn,
                                    const float* __restrict__ vn, float* __restrict__ part) {
  const int bh = blockIdx.x >> 2;       // 0..255
  const int s  = blockIdx.x & 3;
  const int b = bh >> 5, h = bh & 31;
  const int t = threadIdx.x;            // 256
  __shared__ float sQ[DD];
  __shared__ float sc[LC + 1];
  __shared__ float red[256];
  if (t < DD) sQ[t] = q[(size_t)bh * DD + t];
  __syncthreads();
  const int j0 = s * LC;
  const bool last = (s == SPLITS - 1);
  const int nsc = LC + (last ? 1 : 0);
  const float scl = 0.08838834764831845f;   // 1/sqrt(128)

  for (int jj = t; jj < LC; jj += 256) {
    const float4* kp = (const float4*)(kc + ((((size_t)b * LL) + j0 + jj) * HH + h) * DD);
    float acc = 0.f;
#pragma unroll 8
    for (int d4 = 0; d4 < 32; ++d4) {
      float4 kvv = kp[d4];
      acc += sQ[d4*4+0]*kvv.x + sQ[d4*4+1]*kvv.y + sQ[d4*4+2]*kvv.z + sQ[d4*4+3]*kvv.w;
    }
    sc[jj] = acc * scl;
  }
  if (last && t == 0) {
    const float4* kp = (const float4*)(kn + (size_t)bh * DD);
    float acc = 0.f;
#pragma unroll 8
    for (int d4 = 0; d4 < 32; ++d4) {
      float4 kvv = kp[d4];
      acc += sQ[d4*4+0]*kvv.x + sQ[d4*4+1]*kvv.y + sQ[d4*4+2]*kvv.z + sQ[d4*4+3]*kvv.w;
    }
    sc[LC] = acc * scl;
  }
  __syncthreads();

  float lm = -3.4e38f;
  for (int jj = t; jj < nsc; jj += 256) lm = fmaxf(lm, sc[jj]);
  red[t] = lm; __syncthreads();
  for (int o = 128; o > 0; o >>= 1) { if (t < o) red[t] = fmaxf(red[t], red[t + o]); __syncthreads(); }
  const float M = red[0]; __syncthreads();

  float ls = 0.f;
  for (int jj = t; jj < nsc; jj += 256) { float p = __expf(sc[jj] - M); sc[jj] = p; ls += p; }
  red[t] = ls; __syncthreads();
  for (int o = 128; o > 0; o >>= 1) { if (t < o) red[t] += red[t + o]; __syncthreads(); }
  const float S = red[0]; __syncthreads();

  const int d = t & 127, jpar = t >> 7;
  float acc = 0.f;
  for (int j = jpar; j < LC; j += 2)
    acc += sc[j] * vc[((((size_t)b * LL) + j0 + j) * HH + h) * DD + d];
  if (last && jpar == 1) acc += sc[LC] * vn[(size_t)bh * DD + d];
  red[t] = acc; __syncthreads();
  if (t < 128) {
    float* pp = part + ((size_t)bh * SPLITS + s) * 130;
    pp[2 + t] = red[t] + red[t + 128];
    if (t == 0) { pp[0] = M; pp[1] = S; }
  }
}

__global__ void attn_combine_kernel(const float* __restrict__ part, float* __restrict__ ctx_pad) {
  const int b = blockIdx.x >> 5;        // 0..15 (grid 16*32)
  const int h = blockIdx.x & 31;
  const int d = threadIdx.x;            // 128
  const size_t oi = (size_t)b * HID + h * DD + d;
  if (b >= BB) { ctx_pad[oi] = 0.f; return; }
  const float* pp = part + (size_t)(b * HH + h) * SPLITS * 130;
  float M = -3.4e38f;
  for (int s = 0; s < SPLITS; ++s) M = fmaxf(M, pp[s * 130]);
  float denom = 0.f, acc = 0.f;
  for (int s = 0; s < SPLITS; ++s) {
    const float w = __expf(pp[s * 130] - M);
    denom += w * pp[s * 130 + 1];
    acc   += w * pp[s * 130 + 2 + d];
  }
  ctx_pad[oi] = acc / denom;
}

// ---------------- launch ----------------
extern "C" void kernel_launch(void* const* d_in, const int* in_sizes, int n_in,
                              void* d_out, int out_size, void* d_ws, size_t ws_size,
                              hipStream_t stream) {
  const float* hidden  = (const float*)d_in[0];
  const float* k_cache = (const float*)d_in[1];
  const float* v_cache = (const float*)d_in[2];
  const float* w_pack  = (const float*)d_in[3];
  const float* o_proj  = (const float*)d_in[4];
  const float* gate_w  = (const float*)d_in[5];
  const float* up_w    = (const float*)d_in[6];
  const float* down_w  = (const float*)d_in[7];
  const float* ln1_w   = (const float*)d_in[8];
  const float* ln2_w   = (const float*)d_in[9];
  const int*   pos     = (const int*)d_in[10];
  float* out = (float*)d_out;
  float* ws  = (float*)d_ws;

  float* hnorm = ws + OFF_HNORM;
  float* qkv   = ws + OFF_QKV;
  float* qr    = ws + OFF_QR;
  float* kn    = ws + OFF_KN;
  float* vn    = ws + OFF_VN;
  float* part  = ws + OFF_PART;
  float* ctx   = ws + OFF_CTX;
  float* h1    = ws + OFF_H1;
  float* h2    = ws + OFF_H2;
  float* act   = ws + OFF_ACT;
  float* arena = ws + OFF_ARENA;                 // aliased split-K partials
  float* gpart = arena;
  float* upart = arena + (size_t)2 * 16 * INTER; // 352256

  // 1) rmsnorm(hidden) -> 16-row padded
  rmsnorm16_kernel<<<16, 256, 0, stream>>>(hidden, ln1_w, hnorm);

  // 2) qkv = hnorm @ w_pack.T   (N=12288, SK=2 -> 1536 waves)
  gemm16_splitk_kernel<<<192, 256, 0, stream>>>(hnorm, w_pack, arena, HID, 3 * HID, 2);
  reduce_plain_kernel<<<768, 256, 0, stream>>>(arena, qkv, 3 * HID, 2);

  // 3) rope + split
  rope_kernel<<<BB * HH, 128, 0, stream>>>(qkv, pos, qr, kn, vn);

  // 4) attention partials (1024 WGs stream the 1GB KV cache)
  attn_partial_kernel<<<BB * HH * SPLITS, 256, 0, stream>>>(qr, k_cache, v_cache, kn, vn, part);

  // 5) combine -> padded ctx
  attn_combine_kernel<<<16 * HH, 128, 0, stream>>>(part, ctx);

  // 6) h1 = hidden + ctx @ o_proj.T   (N=4096, SK=4 -> 1024 waves)
  gemm16_splitk_kernel<<<128, 256, 0, stream>>>(ctx, o_proj, arena, HID, HID, 4);
  reduce_resid_kernel<<<128, 256, 0, stream>>>(arena, hidden, h1, HID, 4);

  // 7) h2 = rmsnorm(h1)
  rmsnorm16_kernel<<<16, 256, 0, stream>>>(h1, ln2_w, h2);

  // 8) act = silu(h2 @ gate.T) * (h2 @ up.T)   (N=11008, SK=2 -> 1376 waves)
  gemm16_gateup_splitk_kernel<<<172, 256, 0, stream>>>(h2, gate_w, up_w, gpart, upart,
                                                       HID, INTER, 2);
  reduce_gateup_kernel<<<688, 256, 0, stream>>>(gpart, upart, act, INTER, 2);

  // 9) out = h1 + act @ down.T   (K=11008, N=4096, SK=4 -> 1024 waves)
  gemm16_splitk_kernel<<<128, 256, 0, stream>>>(act, down_w, arena, INTER, HID, 4);
  reduce_resid_kernel<<<128, 256, 0, stream>>>(arena, h1, out, HID, 4);
}